// GConvNet_60215441490239
// MI455X (gfx1250) — compile-verified
//
#include <hip/hip_runtime.h>
#include <stdint.h>

// ---------------------------------------------------------------------------
// kNN (k=49, N=16384 nodes, M=32768 targets) + SPH gaussian gather, gfx1250.
// Ranking keys generated with V_WMMA_F32_16X16X4_F32:
//   D[node m][target n] = q_m - 2*(x_n*xn_m + y_n*yn_m)   (monotone in d2)
// A (16x4, M=nodes): row m = (xn, yn, q, pad)
// B (4x16, N=targets): col n = (-2x, -2y, 1, 0)
// D layout: lane owns target n = lane%16; node rows striped over 8 VGPRs
// (lanes 0-15 -> m=0..7, lanes 16-31 -> m=8..15). Per-lane 49-max-heap in LDS.
// Fast path: software-pipelined ds_load_2addr_b64 pairs (next group's A
// operands load while current group's 4 WMMA results are scanned), so the
// dscnt wait is covered by ~40 instructions of independent work.
// Cold path (heap insert, ~k*ln(N/k)=285 accepts/target) is __noinline__.
// ---------------------------------------------------------------------------

typedef float v2f __attribute__((ext_vector_type(2)));
typedef float v8f __attribute__((ext_vector_type(8)));

#define KNN 49
#define CHUNK 512           // nodes staged per pass: float4 -> 8 KB LDS
#define TPB 128             // 4 waves
#define TGT_PER_BLOCK 64    // 4 waves * 16 targets

// monotone (totally ordered) uint encoding of float
__device__ __forceinline__ unsigned enc_key(float f) {
  unsigned b = __float_as_uint(f);
  return b ^ ((b & 0x80000000u) ? 0xFFFFFFFFu : 0x80000000u);
}
__device__ __forceinline__ float dec_key(unsigned e) {
  unsigned b = e ^ ((e & 0x80000000u) ? 0x80000000u : 0xFFFFFFFFu);
  return __uint_as_float(b);
}

// COLD: replace root of 49-entry max-heap with v, sift down; return new root.
// noinline: emitted once instead of ~33 inlined copies of the sift loop.
__device__ __noinline__ unsigned long long heap_replace(
    unsigned long long* __restrict__ H, unsigned long long v) {
  int i = 0;
  for (;;) {
    int l = 2 * i + 1;
    if (l >= KNN) break;
    unsigned long long hc = H[l];
    int c = l;
    if (l + 1 < KNN) {
      unsigned long long hr = H[l + 1];
      if (hr > hc) { c = l + 1; hc = hr; }
    }
    if (hc > v) { H[i] = hc; i = c; } else break;
  }
  H[i] = v;
  return H[0];
}

// scan one 16-node WMMA result tile against this lane's heap
__device__ __forceinline__ void scan_tile(const v8f& d, int jbase,
                                          unsigned long long* __restrict__ H,
                                          unsigned long long& rootU,
                                          float& rootF) {
  float m8 = fminf(fminf(fminf(d[0], d[1]), fminf(d[2], d[3])),
                   fminf(fminf(d[4], d[5]), fminf(d[6], d[7])));
  if (m8 < rootF) {
#pragma unroll
    for (int r = 0; r < 8; ++r) {
      const float key = d[r];
      if (key < rootF) {
        const unsigned long long v =
            ((unsigned long long)enc_key(key) << 32) | (unsigned)(jbase + r);
        rootU = heap_replace(H, v);
        rootF = dec_key((unsigned)(rootU >> 32));
      }
    }
  }
}

__global__ __launch_bounds__(TPB) void knn_sph_wmma(
    const float* __restrict__ x, const float* __restrict__ y,
    const float* __restrict__ nodes, const float* __restrict__ hh,
    const float* __restrict__ uu, float* __restrict__ out, int N) {
  __shared__ float4 sn4[CHUNK];                   // (xn, yn, q, pad), 8 KB
  __shared__ unsigned long long heap[TPB * KNN];  // per-lane heaps, 50 KB

  const int tid = threadIdx.x;
  const int lane = tid & 31;
  const int wave = tid >> 5;
  const int lhalf = lane & 15;
  const bool hi = lane >= 16;

  // lanes L and L+16 both serve target lhalf (disjoint node rows)
  const int t = blockIdx.x * TGT_PER_BLOCK + wave * 16 + lhalf;
  const float tx = x[t];
  const float ty = y[t];

  // B operand (K x 16 targets), loop-invariant:
  //   lanes 0..15 supply K=0,1 : (-2x, -2y); lanes 16..31 supply K=2,3 : (1,0)
  v2f B;
  B.x = hi ? 1.0f : -2.0f * tx;
  B.y = hi ? 0.0f : -2.0f * ty;

  unsigned long long* H = &heap[tid * KNN];
  for (int i = 0; i < KNN; ++i) H[i] = 0xFFFFFFFFFFFFFFFFull;  // sentinels
  unsigned long long rootU = 0xFFFFFFFFFFFFFFFFull;
  float rootF = __builtin_inff();

  const v8f czero = {0.f, 0.f, 0.f, 0.f, 0.f, 0.f, 0.f, 0.f};

  // per-lane element offset into a node record: lo -> (xn,yn), hi -> (q,pad)
  const float* snf = (const float*)sn4;
  const int laneOff = lhalf * 4 + (hi ? 2 : 0);
  const int mhalf = hi ? 8 : 0;

  for (int c0 = 0; c0 < N; c0 += CHUNK) {
    __syncthreads();
    // cooperative stage: (xn, yn, xn^2+yn^2, 0) -> LDS, one b128 store/node
    for (int n = tid; n < CHUNK; n += TPB) {
      const float2 p = ((const float2*)nodes)[c0 + n];
      sn4[n] = make_float4(p.x, p.y, p.x * p.x + p.y * p.y, 0.0f);
    }
    __syncthreads();

    // warm next chunk (bounded: CHUNK*8 bytes) while scanning this one
    if ((c0 + CHUNK < N) && (tid * 64 < CHUNK * 8)) {
      const char* nb = (const char*)(nodes + 2 * (size_t)(c0 + CHUNK));
      __builtin_prefetch(nb + tid * 64, 0, 0);
    }

    // software-pipelined scan: 4 tiles (64 nodes) per iteration
    v2f A0 = *(const v2f*)&snf[0 * 64 + laneOff];
    v2f A1 = *(const v2f*)&snf[1 * 64 + laneOff];
    v2f A2 = *(const v2f*)&snf[2 * 64 + laneOff];
    v2f A3 = *(const v2f*)&snf[3 * 64 + laneOff];
    for (int tile = 0; tile < CHUNK; tile += 64) {
      const v8f d0 = __builtin_amdgcn_wmma_f32_16x16x4_f32(
          false, A0, false, B, (short)0, czero, false, false);
      const v8f d1 = __builtin_amdgcn_wmma_f32_16x16x4_f32(
          false, A1, false, B, (short)0, czero, false, false);
      const v8f d2 = __builtin_amdgcn_wmma_f32_16x16x4_f32(
          false, A2, false, B, (short)0, czero, false, false);
      const v8f d3 = __builtin_amdgcn_wmma_f32_16x16x4_f32(
          false, A3, false, B, (short)0, czero, false, false);

      if (tile + 64 < CHUNK) {  // uniform; loads overlap the scans below
        const int nb = (tile + 64) * 4;
        A0 = *(const v2f*)&snf[nb + 0 * 64 + laneOff];
        A1 = *(const v2f*)&snf[nb + 1 * 64 + laneOff];
        A2 = *(const v2f*)&snf[nb + 2 * 64 + laneOff];
        A3 = *(const v2f*)&snf[nb + 3 * 64 + laneOff];
      }

      const int jb = c0 + tile + mhalf;
      scan_tile(d0, jb + 0, H, rootU, rootF);
      scan_tile(d1, jb + 16, H, rootU, rootF);
      scan_tile(d2, jb + 32, H, rootU, rootF);
      scan_tile(d3, jb + 48, H, rootU, rootF);
    }
  }

  __syncthreads();

  if (!hi) {
    // merge partner half-lane's heap: top-49 of union of two top-49 sets
    const unsigned long long* Hp = &heap[(tid + 16) * KNN];
    for (int q = 0; q < KNN; ++q) {
      const unsigned long long e = Hp[q];
      if (e < rootU) rootU = heap_replace(H, e);
    }
    // SPH gather over exact neighbor set, reference-exact formula
    float dnr = 0.f, nr = 0.f;
    for (int q = 0; q < KNN; ++q) {
      const unsigned j = (unsigned)H[q];
      const float2 p = ((const float2*)nodes)[j];
      const float hj = hh[j];
      const float uj = uu[j];
      const float xh = (tx - p.x) / hj;
      const float yh = (ty - p.y) / hj;
      const float a = expf(-(xh * xh + yh * yh));
      dnr += a;
      nr += uj * a;
    }
    out[t] = nr / dnr;
  }
}

extern "C" void kernel_launch(void* const* d_in, const int* in_sizes, int n_in,
                              void* d_out, int out_size, void* d_ws,
                              size_t ws_size, hipStream_t stream) {
  const float* x = (const float*)d_in[0];
  const float* y = (const float*)d_in[1];
  const float* nodes = (const float*)d_in[2];
  const float* h = (const float*)d_in[3];
  const float* u = (const float*)d_in[4];
  // d_in[5] is k (=49), compile-time constant KNN here
  float* out = (float*)d_out;
  const int M = in_sizes[0];
  const int N = in_sizes[3];
  dim3 grid(M / TGT_PER_BLOCK);
  dim3 block(TPB);
  hipLaunchKernelGGL(knn_sph_wmma, grid, block, 0, stream, x, y, nodes, h, u,
                     out, N);
}